// DiscriminativeLoss_52673478918524
// MI455X (gfx1250) — compile-verified
//
#include <hip/hip_runtime.h>

// Problem constants (match reference setup_inputs)
#define B_  8
#define N_  131072
#define D_  32
#define L_  33
#define PPB 2048            // points per block per streaming kernel
#define CPB (N_ / PPB)      // 64 chunks per batch

typedef __attribute__((ext_vector_type(2))) float v2f;
typedef __attribute__((ext_vector_type(8))) float v8f;

// Workspace layout (floats)
#define OFF_SUMS   0                          // B*33*32
#define OFF_CNT    (OFF_SUMS  + B_*L_*D_)     // B*33
#define OFF_MEANS  (OFF_CNT   + B_*L_)        // B*33*32  (byte offset 34848 = 16B aligned)
#define OFF_HINGE  (OFF_MEANS + B_*L_*D_)     // B*33
#define OFF_PUSH   (OFF_HINGE + B_*L_)        // B
#define WS_FLOATS  (OFF_PUSH  + B_)

__device__ __forceinline__ void gAdd(float* p, float v) {
  __hip_atomic_fetch_add(p, v, __ATOMIC_RELAXED, __HIP_MEMORY_SCOPE_AGENT);
}
__device__ __forceinline__ void sAdd(float* p, float v) {
  __hip_atomic_fetch_add(p, v, __ATOMIC_RELAXED, __HIP_MEMORY_SCOPE_WORKGROUP);
}
__device__ __forceinline__ float waveSum32(float v) {
#pragma unroll
  for (int off = 16; off >= 1; off >>= 1) v += __shfl_xor(v, off, 32);
  return v;
}

// ---------------- Kernel 0: zero workspace ----------------
__global__ void k_zero_ws(float* __restrict__ W) {
  int i = blockIdx.x * blockDim.x + threadIdx.x;
  if (i < WS_FLOATS) W[i] = 0.0f;
}

// ---------------- Kernel 1: segmented sums + counts ----------------
// grid (CPB, B), block 256 (8 waves). Each wave handles 4 points per iteration:
// lane l owns dims 4*(l&7)..+3 of point n + (l>>3). One contiguous 512B
// global_load_b128 per wave per 4 points; 4 ds_add_f32 per lane (1 per dim).
__global__ void k_seg_sums(const float* __restrict__ emb,
                           const int* __restrict__ labels,
                           float* __restrict__ W) {
  __shared__ float s_acc[L_ * D_];
  __shared__ float s_cnt[L_];
  const int tid  = threadIdx.x;
  const int lane = tid & 31;
  const int wave = tid >> 5;
  for (int i = tid; i < L_ * D_; i += 256) s_acc[i] = 0.0f;
  if (tid < L_) s_cnt[tid] = 0.0f;
  __syncthreads();

  const int b  = blockIdx.y;
  const int n0 = blockIdx.x * PPB;
#pragma unroll 2
  for (int n = n0 + wave * 4; n < n0 + PPB; n += 32) {
    const int lab = labels[b * N_ + n + (lane >> 3)];
    const float4 v = *(const float4*)&emb[((size_t)(b * N_ + n)) * D_ + lane * 4];
    float* dst = &s_acc[lab * D_ + (lane & 7) * 4];
    sAdd(&dst[0], v.x);
    sAdd(&dst[1], v.y);
    sAdd(&dst[2], v.z);
    sAdd(&dst[3], v.w);
    if ((lane & 7) == 0) sAdd(&s_cnt[lab], 1.0f);
  }
  __syncthreads();

  float* sums = W + OFF_SUMS + b * L_ * D_;
  float* cnts = W + OFF_CNT  + b * L_;
  for (int i = tid; i < L_ * D_; i += 256) gAdd(&sums[i], s_acc[i]);
  if (tid < L_) gAdd(&cnts[tid], s_cnt[tid]);
}

// ---------------- Kernel 2: means + push term (WMMA f32 Gram) ----------------
// grid B, block 32 (one wave; WMMA needs full EXEC).
__global__ void k_means_push(float* __restrict__ W) {
  __shared__ float s_mean[L_ * D_];
  __shared__ float s_mn[32 * 32];   // normalized means, row r <-> label r+1
  __shared__ float s_g[32 * 32];    // Gram = mn * mn^T
  __shared__ float s_pres[32];

  const int lane = threadIdx.x;
  const int b    = blockIdx.x;
  const float* sums = W + OFF_SUMS + b * L_ * D_;
  const float* cnts = W + OFF_CNT  + b * L_;
  float* means      = W + OFF_MEANS + b * L_ * D_;

  for (int l = 0; l < L_; ++l) {
    const float c = cnts[l];
    const float m = sums[l * D_ + lane] / fmaxf(c, 1.0f);
    s_mean[l * D_ + lane] = m;
    means[l * D_ + lane]  = m;
  }
  // normalize rows for labels 1..32
  for (int r = 0; r < 32; ++r) {
    const float m  = s_mean[(r + 1) * D_ + lane];
    const float ss = waveSum32(m * m);
    const float inv = 1.0f / fmaxf(sqrtf(ss), 1e-12f);
    s_mn[r * 32 + lane] = m * inv;
  }
  s_pres[lane] = (cnts[1 + lane] > 0.0f) ? 1.0f : 0.0f;
  __syncthreads();

  // Gram 32x32 = mn * mn^T via V_WMMA_F32_16X16X4_F32 (4 tiles, K=32 in steps of 4)
  const int mrow  = lane & 15;
  const int kbase = (lane >> 4) * 2;  // VGPR0 = K0/K2, VGPR1 = K1/K3
#pragma unroll
  for (int mi = 0; mi < 32; mi += 16) {
#pragma unroll
    for (int nj = 0; nj < 32; nj += 16) {
      v8f acc = {};
      const int ra = mi + mrow;   // A row (M)
      const int rb = nj + mrow;   // B column (N) -> row of mn (Gram symmetry)
#pragma unroll
      for (int k = 0; k < 32; k += 4) {
        v2f a, bv;
        a.x  = s_mn[ra * 32 + k + kbase];
        a.y  = s_mn[ra * 32 + k + kbase + 1];
        bv.x = s_mn[rb * 32 + k + kbase];
        bv.y = s_mn[rb * 32 + k + kbase + 1];
        acc = __builtin_amdgcn_wmma_f32_16x16x4_f32(
            false, a, false, bv, (short)0, acc, false, false);
      }
#pragma unroll
      for (int v = 0; v < 8; ++v)
        s_g[(mi + v + 8 * (lane >> 4)) * 32 + (nj + mrow)] = acc[v];
    }
  }
  __syncthreads();

  // push: lane = row i; iterate j > i (strict upper triangle)
  const float gii = s_g[lane * 32 + lane];
  float hsum = 0.0f, msum = 0.0f;
  for (int j = lane + 1; j < 32; ++j) {
    const float pm = s_pres[lane] * s_pres[j];
    const float sq = gii + s_g[j * 32 + j] - 2.0f * s_g[lane * 32 + j];
    const float dm = sqrtf(sq + 1e-24f);
    hsum += fmaxf(1.0f - dm, 0.0f) * pm;  // 2*DELTA_D = 1.0
    msum += pm;
  }
  const float ni = waveSum32(s_pres[lane]);
  hsum = waveSum32(hsum);
  msum = waveSum32(msum);
  if (lane == 0)
    W[OFF_PUSH + b] = (ni > 1.0f) ? (hsum / (msum + 1e-6f)) : 0.0f;
}

// ---------------- Kernel 3: per-point hinge distances ----------------
// Same 4-points-per-wave vectorized layout; distance reduction is 4 in-lane
// FMAs + 3-step shfl_xor over the 8-lane group that owns the point.
__global__ void k_hinge(const float* __restrict__ emb,
                        const int* __restrict__ labels,
                        float* __restrict__ W) {
  __shared__ float s_h[L_];
  const int tid  = threadIdx.x;
  const int lane = tid & 31;
  const int wave = tid >> 5;
  if (tid < L_) s_h[tid] = 0.0f;
  __syncthreads();

  const int b  = blockIdx.y;
  const int n0 = blockIdx.x * PPB;
  const float* means = W + OFF_MEANS + b * L_ * D_;
#pragma unroll 2
  for (int n = n0 + wave * 4; n < n0 + PPB; n += 32) {
    const int lab = labels[b * N_ + n + (lane >> 3)];
    const float4 e = *(const float4*)&emb[((size_t)(b * N_ + n)) * D_ + lane * 4];
    const float4 m = *(const float4*)&means[lab * D_ + (lane & 7) * 4];
    const float dx = e.x - m.x, dy = e.y - m.y, dz = e.z - m.z, dw = e.w - m.w;
    float s = dx * dx + dy * dy + dz * dz + dw * dw;
#pragma unroll
    for (int off = 1; off < 8; off <<= 1) s += __shfl_xor(s, off, 32);
    if ((lane & 7) == 0) {
      const float dist = sqrtf(s + 1e-24f);
      sAdd(&s_h[lab], fmaxf(dist - 0.1f, 0.0f));  // DELTA_V = 0.1
    }
  }
  __syncthreads();
  if (tid < L_) gAdd(&W[OFF_HINGE + b * L_ + tid], s_h[tid]);
}

// ---------------- Kernel 4: finalize (3 scalars) ----------------
__global__ void k_finalize(const float* __restrict__ W, float* __restrict__ out) {
  __shared__ float s_pull[B_];
  const int tid  = threadIdx.x;
  const int b    = tid >> 5;       // wave w <-> batch w
  const int li   = 1 + (tid & 31); // labels 1..32
  const float c  = W[OFF_CNT + b * L_ + li];
  const float hs = W[OFF_HINGE + b * L_ + li];
  const float seg  = hs / fmaxf(c, 1.0f);
  const float pres = (c > 0.0f) ? 1.0f : 0.0f;
  const float ssum = waveSum32(seg);
  const float ni   = waveSum32(pres);
  if ((tid & 31) == 0) s_pull[b] = ssum / (ni + 1e-6f);
  __syncthreads();
  if (tid == 0) {
    float pull = 0.0f, push = 0.0f;
    for (int i = 0; i < B_; ++i) { pull += s_pull[i]; push += W[OFF_PUSH + i]; }
    pull /= (float)B_;
    push /= (float)B_;
    out[0] = pull + push;
    out[1] = pull;
    out[2] = push;
  }
}

extern "C" void kernel_launch(void* const* d_in, const int* in_sizes, int n_in,
                              void* d_out, int out_size, void* d_ws, size_t ws_size,
                              hipStream_t stream) {
  const float* emb = (const float*)d_in[0];
  const int*   lab = (const int*)d_in[1];
  float*       out = (float*)d_out;
  float*       W   = (float*)d_ws;
  (void)in_sizes; (void)n_in; (void)out_size; (void)ws_size;

  k_zero_ws   <<<(WS_FLOATS + 255) / 256, 256, 0, stream>>>(W);
  k_seg_sums  <<<dim3(CPB, B_), 256, 0, stream>>>(emb, lab, W);
  k_means_push<<<B_, 32, 0, stream>>>(W);
  k_hinge     <<<dim3(CPB, B_), 256, 0, stream>>>(emb, lab, W);
  k_finalize  <<<1, 256, 0, stream>>>(W, out);
}